// OpenseekCDMoE_58892591562979
// MI455X (gfx1250) — compile-verified
//
#include <hip/hip_runtime.h>
#include <hip/hip_bf16.h>
#include <math.h>
#include <stdint.h>

#define HIDDEN 2048
#define INTER  8192
#define RETR   128
#define NUM_KEYS 64
#define TOPK   8
#define NTOK   8192   // B*S = 4*2048

typedef float v2f __attribute__((ext_vector_type(2)));
typedef float v8f __attribute__((ext_vector_type(8)));

__device__ __forceinline__ v8f wmma4(v2f a, v2f b, v8f c) {
    // D = A(16x4 f32) * B(4x16 f32) + C(16x16 f32)
    return __builtin_amdgcn_wmma_f32_16x16x4_f32(
        /*neg_a=*/false, a, /*neg_b=*/false, b,
        /*c_mod=*/(short)0, c, /*reuse_a=*/false, /*reuse_b=*/false);
}

// Async copy: global -> LDS, 16B per lane, tracked by ASYNCcnt.
// GVS mode: mem_addr = SGPR base + VGPR byte-offset. LDS dest address is the
// low 32 bits of the flat shared pointer (LDS aperture rule: addr[31:0]).
__device__ __forceinline__ void async_b128(void* lds_ptr, const float* base,
                                           uint32_t byte_off) {
    uint32_t lds_off = (uint32_t)(uintptr_t)lds_ptr;
    asm volatile("global_load_async_to_lds_b128 %0, %1, %2"
                 :: "v"(lds_off), "v"(byte_off), "s"(base)
                 : "memory");
}
__device__ __forceinline__ void wait_async0() {
    asm volatile("s_wait_asynccnt 0x0" ::: "memory");
}

// ---------------------------------------------------------------------------
// 1) Q = X @ W_q   [NTOK x RETR]
// ---------------------------------------------------------------------------
__global__ __launch_bounds__(RETR) void queries_kernel(
        const float* __restrict__ X, const float* __restrict__ Wq,
        float* __restrict__ Q) {
    const int i = blockIdx.x;      // token
    const int r = threadIdx.x;     // retrieval dim
    const float* xr = X + (size_t)i * HIDDEN;
    float acc = 0.f;
    for (int h = 0; h < HIDDEN; ++h)
        acc = fmaf(xr[h], Wq[(size_t)h * RETR + r], acc);
    Q[(size_t)i * RETR + r] = acc;
}

// ---------------------------------------------------------------------------
// 2) product-key routing + two-level stable top-8
// ---------------------------------------------------------------------------
// stable descending top-8 insertion: ties keep the earlier index (matches
// jax.lax.top_k). One bubble pass after replacing the tail; fully unrolled
// so the arrays stay in registers.
__device__ __forceinline__ void topk_update(float v, int id,
                                            float vals[TOPK], int ids[TOPK]) {
    if (v <= vals[TOPK - 1]) return;          // strict: equal keeps old (earlier) id
    vals[TOPK - 1] = v; ids[TOPK - 1] = id;
#pragma unroll
    for (int p = TOPK - 1; p > 0; --p) {
        if (vals[p] > vals[p - 1]) {          // strict: stable on ties
            float tv = vals[p]; vals[p] = vals[p - 1]; vals[p - 1] = tv;
            int   ti = ids[p];  ids[p]  = ids[p - 1];  ids[p - 1]  = ti;
        }
    }
}

__global__ __launch_bounds__(256) void routing_kernel(
        const float* __restrict__ Q, const float* __restrict__ keys,
        float* __restrict__ scores, int* __restrict__ indices) {
    const int j = blockIdx.x * blockDim.x + threadIdx.x;
    if (j >= NTOK) return;
    // view(2, N, 64) of the [N,128] query matrix:
    //   t=0 row j -> token j/2,        half j%2
    //   t=1 row j -> token NTOK/2+j/2, half j%2
    const float* q0 = Q + (size_t)(j >> 1) * RETR + (j & 1) * (RETR / 2);
    const float* q1 = Q + (size_t)((NTOK >> 1) + (j >> 1)) * RETR + (j & 1) * (RETR / 2);

    float sx[TOPK], sy[TOPK]; int ix[TOPK], iy[TOPK];
#pragma unroll
    for (int t = 0; t < TOPK; ++t) { sx[t] = -INFINITY; sy[t] = -INFINITY; ix[t] = 0; iy[t] = 0; }

    for (int k = 0; k < NUM_KEYS; ++k) {
        float r0 = 0.f, r1 = 0.f;
        for (int d = 0; d < RETR / 2; ++d) {
            r0 = fmaf(q0[d], keys[d * NUM_KEYS + k], r0);
            r1 = fmaf(q1[d], keys[(RETR / 2) * NUM_KEYS + d * NUM_KEYS + k], r1);
        }
        topk_update(r0, k, sx, ix);
        topk_update(r1, k, sy, iy);
    }

    float cv[TOPK]; int cid[TOPK];
#pragma unroll
    for (int t = 0; t < TOPK; ++t) { cv[t] = -INFINITY; cid[t] = 0; }
    // iterate in flat (a*8+b) order so stability matches jax top_k positions
    for (int a = 0; a < TOPK; ++a)
        for (int b = 0; b < TOPK; ++b)
            topk_update(sx[a] + sy[b], ix[a] * NUM_KEYS + iy[b], cv, cid);

#pragma unroll
    for (int t = 0; t < TOPK; ++t) {
        scores[(size_t)j * TOPK + t]  = cv[t];
        indices[(size_t)j * TOPK + t] = cid[t];
    }
}

// ---------------------------------------------------------------------------
// 3) expert branch: out = sum_k silu(x.de_k)*softmax(s)_k * ue_k  (per token)
//    embed tables are 67MB total -> L2-resident gathers.
// ---------------------------------------------------------------------------
__global__ __launch_bounds__(256) void expert_kernel(
        const float* __restrict__ X, const float* __restrict__ down_embed,
        const float* __restrict__ up_embed, const float* __restrict__ scores,
        const int* __restrict__ indices, float* __restrict__ out) {
    __shared__ float part[256 * TOPK];
    __shared__ float ewsh[TOPK];
    __shared__ float wsh[TOPK];
    const int i = blockIdx.x;
    const int t = threadIdx.x;

    int idx[TOPK];
#pragma unroll
    for (int k = 0; k < TOPK; ++k) idx[k] = indices[(size_t)i * TOPK + k];

    const float* xr = X + (size_t)i * HIDDEN;
    float p[TOPK];
#pragma unroll
    for (int k = 0; k < TOPK; ++k) p[k] = 0.f;

    for (int h = t; h < HIDDEN; h += 256) {
        float xv = xr[h];
#pragma unroll
        for (int k = 0; k < TOPK; ++k)
            p[k] = fmaf(xv, down_embed[(size_t)idx[k] * HIDDEN + h], p[k]);
    }
#pragma unroll
    for (int k = 0; k < TOPK; ++k) part[t * TOPK + k] = p[k];
    __syncthreads();

    if (t < TOPK) {
        float s = 0.f;
        for (int u = 0; u < 256; ++u) s += part[u * TOPK + t];
        ewsh[t] = s;
    }
    __syncthreads();

    if (t == 0) {
        float mx = -INFINITY;
#pragma unroll
        for (int k = 0; k < TOPK; ++k) mx = fmaxf(mx, scores[(size_t)i * TOPK + k]);
        float e[TOPK]; float den = 0.f;
#pragma unroll
        for (int k = 0; k < TOPK; ++k) { e[k] = __expf(scores[(size_t)i * TOPK + k] - mx); den += e[k]; }
#pragma unroll
        for (int k = 0; k < TOPK; ++k) {
            float ew  = ewsh[k];
            float sil = ew / (1.f + __expf(-ew));
            wsh[k] = sil * (e[k] / den);
        }
    }
    __syncthreads();

    for (int h = t; h < HIDDEN; h += 256) {
        float acc = 0.f;
#pragma unroll
        for (int k = 0; k < TOPK; ++k)
            acc = fmaf(wsh[k], up_embed[(size_t)idx[k] * HIDDEN + h], acc);
        out[(size_t)i * HIDDEN + h] = acc;   // expert states; down GEMM adds MLP on top
    }
}

// ---------------------------------------------------------------------------
// WMMA GEMM tiles: block 128x64, 8 waves, wave tile 32x32 (2x2 of 16x16),
// BK=16 -> 4 chained v_wmma_f32_16x16x4_f32 per accumulator per K-step.
// Tiles staged with global_load_async_to_lds_b128 (ASYNCcnt) and double
// buffered: prefetch tile t+1 while computing tile t.
// A row-major [BM][BK+4]: fragment reads sweep M at fixed K with row stride
// 20 words -> banks 20*m mod 64 all distinct (conflict-free), and each lane's
// (k,k+1) pair is a contiguous ds_load_b64.
// ---------------------------------------------------------------------------
#define BM 128
#define BN 64
#define BK 16
#define APAD 4

// 4) fused gate/up GEMM + SwiGLU epilogue: H = silu(X@Wg) * (X@Wu)
__global__ __launch_bounds__(256) void gateup_kernel(
        const float* __restrict__ X, const float* __restrict__ Wg,
        const float* __restrict__ Wu, float* __restrict__ H) {
    __shared__ float As [2][BM][BK + APAD];
    __shared__ float Bgs[2][BK][BN];
    __shared__ float Bus[2][BK][BN];

    const int t    = threadIdx.x;
    const int lane = t & 31;
    const int wave = t >> 5;
    const int wr   = (wave & 3) * 32;       // wave row offset inside block tile
    const int wc   = (wave >> 2) * 32;      // wave col offset
    const int m    = lane & 15;             // M / N index within 16
    const int ksel = (lane >> 4) * 2;       // lanes 16-31 hold K={2,3}
    const int m0   = blockIdx.y * BM;
    const int n0   = blockIdx.x * BN;

    // per-thread async-load coordinates (16B chunks)
    const int ar0 = t >> 2,            ak0 = (t & 3) * 4;          // A chunk 0
    const int ar1 = (t + 256) >> 2,    ak1 = ((t + 256) & 3) * 4;  // A chunk 1
    const int bk  = t >> 4,            bc  = (t & 15) * 4;         // B chunk

    const v8f vzero = {};
    v8f accg[2][2], accu[2][2];
#pragma unroll
    for (int i = 0; i < 2; ++i)
#pragma unroll
        for (int j = 0; j < 2; ++j) { accg[i][j] = vzero; accu[i][j] = vzero; }

    // ---- stage tile 0 ----
    async_b128(&As[0][ar0][ak0], X, (uint32_t)(((m0 + ar0) * HIDDEN + ak0) * 4));
    async_b128(&As[0][ar1][ak1], X, (uint32_t)(((m0 + ar1) * HIDDEN + ak1) * 4));
    async_b128(&Bgs[0][bk][bc], Wg, (uint32_t)((bk * INTER + n0 + bc) * 4));
    async_b128(&Bus[0][bk][bc], Wu, (uint32_t)((bk * INTER + n0 + bc) * 4));
    wait_async0();
    __syncthreads();

    int cur = 0;
    for (int k0 = 0; k0 < HIDDEN; k0 += BK) {
        const int nxt = cur ^ 1;
        if (k0 + BK < HIDDEN) {   // prefetch next tile into the other buffer
            const int kn = k0 + BK;
            async_b128(&As[nxt][ar0][ak0], X, (uint32_t)(((m0 + ar0) * HIDDEN + kn + ak0) * 4));
            async_b128(&As[nxt][ar1][ak1], X, (uint32_t)(((m0 + ar1) * HIDDEN + kn + ak1) * 4));
            async_b128(&Bgs[nxt][bk][bc], Wg, (uint32_t)(((kn + bk) * INTER + n0 + bc) * 4));
            async_b128(&Bus[nxt][bk][bc], Wu, (uint32_t)(((kn + bk) * INTER + n0 + bc) * 4));
        }

#pragma unroll
        for (int k4 = 0; k4 < BK; k4 += 4) {
            v2f a[2], bg[2], bu[2];
#pragma unroll
            for (int i = 0; i < 2; ++i) {
                a[i].x = As[cur][wr + i * 16 + m][k4 + ksel];
                a[i].y = As[cur][wr + i * 16 + m][k4 + ksel + 1];
            }
#pragma unroll
            for (int j = 0; j < 2; ++j) {
                bg[j].x = Bgs[cur][k4 + ksel    ][wc + j * 16 + m];
                bg[j].y = Bgs[cur][k4 + ksel + 1][wc + j * 16 + m];
                bu[j].x = Bus[cur][k4 + ksel    ][wc + j * 16 + m];
                bu[j].y = Bus[cur][k4 + ksel + 1][wc + j * 16 + m];
            }
#pragma unroll
            for (int i = 0; i < 2; ++i)
#pragma unroll
                for (int j = 0; j < 2; ++j) {
                    accg[i][j] = wmma4(a[i], bg[j], accg[i][j]);
                    accu[i][j] = wmma4(a[i], bu[j], accu[i][j]);
                }
        }

        wait_async0();       // my prefetch writes to LDS are done
        __syncthreads();     // everyone finished reading buffer `cur`
        cur = nxt;
    }

    // C/D layout: VGPR r -> M=r (lanes 0-15) / M=r+8 (lanes 16-31), N=lane%16
    const int rlo = (lane >> 4) * 8;
    const int cn  = lane & 15;
#pragma unroll
    for (int i = 0; i < 2; ++i)
#pragma unroll
        for (int j = 0; j < 2; ++j)
#pragma unroll
            for (int r = 0; r < 8; ++r) {
                float g = accg[i][j][r];
                float u = accu[i][j][r];
                float h = (g / (1.f + __expf(-g))) * u;
                size_t row = (size_t)(m0 + wr + i * 16 + r + rlo);
                size_t col = (size_t)(n0 + wc + j * 16 + cn);
                H[row * INTER + col] = h;
            }
}

// 5) OUT = H @ Wd + expert_states (already in OUT)
__global__ __launch_bounds__(256) void down_kernel(
        const float* __restrict__ H, const float* __restrict__ Wd,
        float* __restrict__ Out) {
    __shared__ float As[2][BM][BK + APAD];
    __shared__ float Bs[2][BK][BN];

    const int t    = threadIdx.x;
    const int lane = t & 31;
    const int wave = t >> 5;
    const int wr   = (wave & 3) * 32;
    const int wc   = (wave >> 2) * 32;
    const int m    = lane & 15;
    const int ksel = (lane >> 4) * 2;
    const int m0   = blockIdx.y * BM;
    const int n0   = blockIdx.x * BN;

    const int ar0 = t >> 2,         ak0 = (t & 3) * 4;
    const int ar1 = (t + 256) >> 2, ak1 = ((t + 256) & 3) * 4;
    const int bk  = t >> 4,         bc  = (t & 15) * 4;

    const v8f vzero = {};
    v8f acc[2][2];
#pragma unroll
    for (int i = 0; i < 2; ++i)
#pragma unroll
        for (int j = 0; j < 2; ++j) acc[i][j] = vzero;

    async_b128(&As[0][ar0][ak0], H, (uint32_t)(((m0 + ar0) * INTER + ak0) * 4));
    async_b128(&As[0][ar1][ak1], H, (uint32_t)(((m0 + ar1) * INTER + ak1) * 4));
    async_b128(&Bs[0][bk][bc], Wd, (uint32_t)((bk * HIDDEN + n0 + bc) * 4));
    wait_async0();
    __syncthreads();

    int cur = 0;
    for (int k0 = 0; k0 < INTER; k0 += BK) {
        const int nxt = cur ^ 1;
        if (k0 + BK < INTER) {
            const int kn = k0 + BK;
            async_b128(&As[nxt][ar0][ak0], H, (uint32_t)(((m0 + ar0) * INTER + kn + ak0) * 4));
            async_b128(&As[nxt][ar1][ak1], H, (uint32_t)(((m0 + ar1) * INTER + kn + ak1) * 4));
            async_b128(&Bs[nxt][bk][bc], Wd, (uint32_t)(((kn + bk) * HIDDEN + n0 + bc) * 4));
        }

#pragma unroll
        for (int k4 = 0; k4 < BK; k4 += 4) {
            v2f a[2], b[2];
#pragma unroll
            for (int i = 0; i < 2; ++i) {
                a[i].x = As[cur][wr + i * 16 + m][k4 + ksel];
                a[i].y = As[cur][wr + i * 16 + m][k4 + ksel + 1];
            }
#pragma unroll
            for (int j = 0; j < 2; ++j) {
                b[j].x = Bs[cur][k4 + ksel    ][wc + j * 16 + m];
                b[j].y = Bs[cur][k4 + ksel + 1][wc + j * 16 + m];
            }
#pragma unroll
            for (int i = 0; i < 2; ++i)
#pragma unroll
                for (int j = 0; j < 2; ++j)
                    acc[i][j] = wmma4(a[i], b[j], acc[i][j]);
        }

        wait_async0();
        __syncthreads();
        cur = nxt;
    }

    const int rlo = (lane >> 4) * 8;
    const int cn  = lane & 15;
#pragma unroll
    for (int i = 0; i < 2; ++i)
#pragma unroll
        for (int j = 0; j < 2; ++j)
#pragma unroll
            for (int r = 0; r < 8; ++r) {
                size_t row = (size_t)(m0 + wr + i * 16 + r + rlo);
                size_t col = (size_t)(n0 + wc + j * 16 + cn);
                size_t off = row * HIDDEN + col;
                Out[off] = Out[off] + acc[i][j][r];   // add MLP onto expert states
            }
}

// ---------------------------------------------------------------------------
extern "C" void kernel_launch(void* const* d_in, const int* in_sizes, int n_in,
                              void* d_out, int out_size, void* d_ws, size_t ws_size,
                              hipStream_t stream) {
    const float* X    = (const float*)d_in[0];  // hidden_states [4,2048,2048]
    const float* Wq   = (const float*)d_in[1];  // [2048,128]
    const float* keys = (const float*)d_in[2];  // [2,64,64]
    const float* dem  = (const float*)d_in[3];  // down_embed [4096,2048]
    const float* uem  = (const float*)d_in[4];  // up_embed   [4096,2048]
    const float* Wg   = (const float*)d_in[5];  // [2048,8192]
    const float* Wu   = (const float*)d_in[6];  // [2048,8192]
    const float* Wd   = (const float*)d_in[7];  // [8192,2048]
    float* out = (float*)d_out;

    // workspace layout
    char* ws = (char*)d_ws;
    float* Q      = (float*)ws;                                   // 4 MB
    char*  p      = ws + (size_t)NTOK * RETR * sizeof(float);
    float* scores = (float*)p;                                    // 256 KB
    p += (size_t)NTOK * TOPK * sizeof(float);
    int*   idx    = (int*)p;                                      // 256 KB
    p += (size_t)NTOK * TOPK * sizeof(int);
    float* H      = (float*)p;                                    // 268 MB

    queries_kernel<<<NTOK, RETR, 0, stream>>>(X, Wq, Q);
    routing_kernel<<<NTOK / 256, 256, 0, stream>>>(Q, keys, scores, idx);
    expert_kernel<<<NTOK, 256, 0, stream>>>(X, dem, uem, scores, idx, out);
    gateup_kernel<<<dim3(INTER / BN, NTOK / BM), 256, 0, stream>>>(X, Wg, Wu, H);
    down_kernel<<<dim3(HIDDEN / BN, NTOK / BM), 256, 0, stream>>>(H, Wd, out);
}